// TensorVMSplit_927712936666
// MI455X (gfx1250) — compile-verified
//
#include <hip/hip_runtime.h>
#include <hip/hip_bf16.h>

typedef __attribute__((ext_vector_type(2))) float v2f;
typedef __attribute__((ext_vector_type(8))) float v8f;

namespace {
constexpr int kGrid   = 300;
constexpr int kPlane  = kGrid * kGrid;   // 90000 floats per channel plane
constexpr int kOutDim = 48;              // APP_DIM == 3*N_COMP == 48
constexpr int kRowPad = 50;              // 48-col feat rows padded -> bank-conflict-free
}

// One wave = one 16-point tile per iteration.
//  Phase 1: gather/sample -> feat[16][48] in per-wave LDS
//  Phase 2: 36x V_WMMA_F32_16X16X4_F32 : out[16x48] = feat[16x48] @ basis_w^T + b
__global__ __launch_bounds__(256)
void tensorvm_split_kernel(const float* __restrict__ xyz,
                           const float* __restrict__ planes,
                           const float* __restrict__ lines,
                           const float* __restrict__ basis_w,
                           const float* __restrict__ basis_b,
                           float* __restrict__ out,
                           int n_points, int n_tiles)
{
    __shared__ float feat[8][16 * kRowPad];   // 8 waves/block, 3200 B each

    const int lane = threadIdx.x & 31;
    const int wid  = threadIdx.x >> 5;
    const int p    = lane & 15;   // point-in-tile (M) and N index
    const int half = lane >> 4;   // selects K pair {0,1} vs {2,3} (wave32 WMMA layout)

    // ---- loop-invariant B operand (basis_w^T, 48x48) held in VGPRs ----
    // B[k][16j+n]: lane<16 -> VGPR0=K(4kk+0),VGPR1=K(4kk+1); lane>=16 -> K(4kk+2),K(4kk+3)
    v2f Bm[3][12];
#pragma unroll
    for (int j = 0; j < 3; ++j) {
#pragma unroll
        for (int kk = 0; kk < 12; ++kk) {
            const int row = 16 * j + p;          // N (output dim)
            const int col = 4 * kk + 2 * half;   // K
            Bm[j][kk] = *(const v2f*)(basis_w + row * kOutDim + col);
        }
    }
    float biasv[3];
#pragma unroll
    for (int j = 0; j < 3; ++j) biasv[j] = basis_b[16 * j + p];

    float* myfeat = &feat[wid][0];

    const int gwave  = blockIdx.x * 8 + wid;
    const int nwaves = gridDim.x * 8;

    for (int tile = gwave; tile < n_tiles; tile += nwaves) {
        const int base = tile * 16;
        int gp = base + p; if (gp > n_points - 1) gp = n_points - 1;

        const float x = xyz[gp * 3 + 0];
        const float y = xyz[gp * 3 + 1];
        const float z = xyz[gp * 3 + 2];

        // Speculative prefetch of next tile's coords (-> global_prefetch_b8).
        __builtin_prefetch(xyz + (long)(base + nwaves * 16 + p) * 3, 0, 0);

        // MAT_MODE=((0,1),(0,2),(1,2))  VEC_MODE=(2,1,0); constant-folded after unroll.
        const float gxs[3] = { x, x, y };
        const float gys[3] = { y, z, z };
        const float gvs[3] = { z, y, x };

        // Lanes 0-15 compute channels 0..23, lanes 16-31 compute 24..47 of point p.
#pragma unroll
        for (int mode = 0; mode < 3; ++mode) {
            const float gx = gxs[mode], gy = gys[mode], gv = gvs[mode];

            const float ix  = (gx + 1.0f) * 0.5f * (float)(kGrid - 1);
            const float iy  = (gy + 1.0f) * 0.5f * (float)(kGrid - 1);
            const float x0f = fminf(fmaxf(floorf(ix), 0.0f), (float)(kGrid - 1));
            const float y0f = fminf(fmaxf(floorf(iy), 0.0f), (float)(kGrid - 1));
            const float wx  = ix - x0f;
            const float wy  = iy - y0f;
            const int   x0  = (int)x0f, y0 = (int)y0f;
            const int   x1  = min(x0 + 1, kGrid - 1);
            const int   y1  = min(y0 + 1, kGrid - 1);

            const float il  = (gv + 1.0f) * 0.5f * (float)(kGrid - 1);
            const float l0f = fminf(fmaxf(floorf(il), 0.0f), (float)(kGrid - 1));
            const float wl  = il - l0f;
            const int   l0  = (int)l0f;
            const int   l1  = min(l0 + 1, kGrid - 1);

            const float w00 = (1.0f - wx) * (1.0f - wy);
            const float w01 = wx * (1.0f - wy);
            const float w10 = (1.0f - wx) * wy;
            const float w11 = wx * wy;
            const float wl0 = 1.0f - wl;

            const int o00 = y0 * kGrid + x0, o01 = y0 * kGrid + x1;
            const int o10 = y1 * kGrid + x0, o11 = y1 * kGrid + x1;

            const float* pl = planes + mode * (16 * kPlane);
            const float* ln = lines  + mode * (16 * kGrid);

            int cbeg = half * 24 - mode * 16;      if (cbeg < 0)  cbeg = 0;
            int cend = half * 24 + 24 - mode * 16; if (cend > 16) cend = 16;
            for (int cc = cbeg; cc < cend; ++cc) {
                const float* pc  = pl + cc * kPlane;       // L2-resident gathers
                const float  v00 = pc[o00], v01 = pc[o01];
                const float  v10 = pc[o10], v11 = pc[o11];
                const float  pv  = v00 * w00 + v01 * w01 + v10 * w10 + v11 * w11;
                const float* lc  = ln + cc * kGrid;
                const float  lv  = lc[l0] * wl0 + lc[l1] * wl;
                myfeat[p * kRowPad + mode * 16 + cc] = pv * lv;
            }
        }

        // Per-wave LDS: drain DS stores before cross-lane reads (no block barrier needed).
        asm volatile("s_wait_dscnt 0x0" ::: "memory");

        // ---- D(16x48) = A(16x48) x B(48x48), K split into 12 chunks of 4 ----
        v8f acc0 = {}, acc1 = {}, acc2 = {};
#pragma unroll
        for (int kk = 0; kk < 12; ++kk) {
            // A 16x4 f32 layout: lanes 0-15 K={4kk,4kk+1}, lanes 16-31 K={4kk+2,4kk+3}
            const v2f A = *(const v2f*)(myfeat + p * kRowPad + 4 * kk + 2 * half);
            acc0 = __builtin_amdgcn_wmma_f32_16x16x4_f32(false, A, false, Bm[0][kk],
                                                         (short)0, acc0, false, false);
            acc1 = __builtin_amdgcn_wmma_f32_16x16x4_f32(false, A, false, Bm[1][kk],
                                                         (short)0, acc1, false, false);
            acc2 = __builtin_amdgcn_wmma_f32_16x16x4_f32(false, A, false, Bm[2][kk],
                                                         (short)0, acc2, false, false);
        }

        // D layout: VGPR r -> row (r + 8*half), col = p (+16j). NT stores: 402 MB
        // output stream must not evict the L2-resident planes.
#pragma unroll
        for (int r = 0; r < 8; ++r) {
            const int row = base + r + half * 8;
            if (row < n_points) {
                float* orow = out + (long)row * kOutDim;
                __builtin_nontemporal_store(acc0[r] + biasv[0], orow +  0 + p);
                __builtin_nontemporal_store(acc1[r] + biasv[1], orow + 16 + p);
                __builtin_nontemporal_store(acc2[r] + biasv[2], orow + 32 + p);
            }
        }
        // WAR on LDS across iterations is safe: same-wave DS ops execute in order,
        // and the A loads already completed (data fed the WMMAs above).
    }
}

extern "C" void kernel_launch(void* const* d_in, const int* in_sizes, int n_in,
                              void* d_out, int out_size, void* d_ws, size_t ws_size,
                              hipStream_t stream) {
    const float* xyz     = (const float*)d_in[0];
    const float* planes  = (const float*)d_in[1];
    const float* lines   = (const float*)d_in[2];
    const float* basis_w = (const float*)d_in[3];
    const float* basis_b = (const float*)d_in[4];
    float* out = (float*)d_out;

    const int n_points = in_sizes[0] / 3;            // 2,097,152
    const int n_tiles  = (n_points + 15) / 16;       // 131,072

    int blocks = (n_tiles + 7) / 8;                  // 8 waves (tiles) per block
    if (blocks > 4096) blocks = 4096;                // grid-stride, 4 tiles/wave

    tensorvm_split_kernel<<<blocks, 256, 0, stream>>>(
        xyz, planes, lines, basis_w, basis_b, out, n_points, n_tiles);
}